// HLocSagePlus_14963666059367
// MI455X (gfx1250) — compile-verified
//
#include <hip/hip_runtime.h>
#include <hip/hip_bf16.h>

#define N_NODES 100000
#define R_REL   4
#define E_EDGES 300000
#define FEAT    64
#define HD      64
#define S_SEEDS 4096
#define P_HEADS 2
#define KC      32       // K-chunk staged in LDS

typedef __attribute__((ext_vector_type(2))) float v2f;
typedef __attribute__((ext_vector_type(8))) float v8f;
typedef __attribute__((ext_vector_type(4))) int   v4i;

#if defined(__has_builtin)
#if __has_builtin(__builtin_amdgcn_global_load_async_to_lds_b128)
#define USE_ASYNC_LDS 1
#endif
#endif

__device__ __forceinline__ void async_wait_all() {
#if defined(USE_ASYNC_LDS)
#if __has_builtin(__builtin_amdgcn_s_wait_asynccnt)
  __builtin_amdgcn_s_wait_asynccnt(0);
#else
  asm volatile("s_wait_asynccnt 0x0" ::: "memory");
#endif
#endif
}

__device__ __forceinline__ void copy16_to_lds(const float* gsrc, float* ldst) {
#if defined(USE_ASYNC_LDS)
  __builtin_amdgcn_global_load_async_to_lds_b128(
      (__attribute__((address_space(1))) v4i*)gsrc,
      (__attribute__((address_space(3))) v4i*)ldst, 0, 0);
#else
  *(float4*)ldst = *(const float4*)gsrc;
#endif
}

// ---------------------------------------------------------------------------
// Tiled fp32 WMMA GEMM: D[z] = act(A[z] @ B[z] + bias[z])
// Block = 128 threads (4 waves) -> 64(M) x 64(N) tile of D.
// B chunk (KC x 64, 8KB) staged in LDS via async-to-LDS; A frags from global.
// Requires N % 64 == 0, K % KC == 0, M % 16 == 0.
// ACT: 0=none, 1=relu, 2=leaky_relu(0.01), 3=tanh
// ---------------------------------------------------------------------------
template<int ACT>
__global__ __launch_bounds__(128) void gemm_wmma_tiled(
    const float* __restrict__ A, const float* __restrict__ B,
    const float* __restrict__ bias, float* __restrict__ D,
    int M, int Nn, int K,
    long sA, long sB, long sBias, long sD)
{
  __shared__ float Bs[KC * 64];          // 8 KB

  const int wave   = threadIdx.x >> 5;
  const int lane   = threadIdx.x & 31;
  const int Mtiles = M >> 4;
  int mt = blockIdx.x * 4 + wave;
  const bool active = (mt < Mtiles);
  if (!active) mt = Mtiles - 1;          // clamp: wave-uniform, keeps barriers/EXEC sane
  const int m0 = mt << 4;
  const int n0 = blockIdx.y << 6;
  const int z  = blockIdx.z;

  const float* Ab = A + (long)z * sA;
  const float* Bb = B + (long)z * sB;

  const int am    = m0 + (lane & 15);
  const int kh    = (lane >> 4) << 1;    // 0 or 2
  const int nlane = lane & 15;

  v8f zero = {0.f,0.f,0.f,0.f,0.f,0.f,0.f,0.f};
  v8f acc[4];
  acc[0] = zero; acc[1] = zero; acc[2] = zero; acc[3] = zero;

  for (int kc = 0; kc < K; kc += KC) {
    // ---- stage B chunk [kc..kc+KC) x [n0..n0+64) into LDS (512 float4) ----
#pragma unroll
    for (int j = 0; j < 4; ++j) {
      const int q   = threadIdx.x + j * 128;   // 0..511
      const int row = q >> 4;                  // 16 float4 per row
      const int c4  = (q & 15) << 2;
      copy16_to_lds(Bb + (long)(kc + row) * Nn + n0 + c4, &Bs[row * 64 + c4]);
    }
    async_wait_all();
    __syncthreads();

    // ---- 8 k-steps x 4 N-tiles of WMMA ----
    const float* ap = Ab + (long)am * K + kc + kh;
#pragma unroll
    for (int kk = 0; kk < KC; kk += 4) {
      v2f a;
      a.x = ap[kk];
      a.y = ap[kk + 1];
      const float* bp0 = &Bs[(kk + kh) * 64 + nlane];
#pragma unroll
      for (int j = 0; j < 4; ++j) {
        v2f b;
        b.x = bp0[j * 16];
        b.y = bp0[64 + j * 16];
        acc[j] = __builtin_amdgcn_wmma_f32_16x16x4_f32(false, a, false, b,
                                                       (short)0, acc[j], false, false);
      }
    }
    __syncthreads();
  }

  if (active) {
    float* Db = D + (long)z * sD;
    const int rbase = m0 + ((lane >> 4) << 3);
#pragma unroll
    for (int j = 0; j < 4; ++j) {
      const int bn = n0 + j * 16 + nlane;
      const float bv = bias ? bias[(long)z * sBias + bn] : 0.f;
#pragma unroll
      for (int v = 0; v < 8; ++v) {
        float x = acc[j][v] + bv;
        if (ACT == 1)      x = fmaxf(x, 0.f);
        else if (ACT == 2) x = (x > 0.f) ? x : 0.01f * x;
        else if (ACT == 3) x = tanhf(x);
        Db[(long)(rbase + v) * Nn + bn] = x;
      }
    }
  }
}

// ---------------------------------------------------------------------------
// Fallback: one wave per 16x16 tile, fully global (used for N=32 GEMM).
// ---------------------------------------------------------------------------
template<int ACT>
__global__ __launch_bounds__(128) void gemm_wmma_small(
    const float* __restrict__ A, const float* __restrict__ B,
    const float* __restrict__ bias, float* __restrict__ D,
    int M, int Nn, int K,
    long sA, long sB, long sBias, long sD)
{
  const int wave   = threadIdx.x >> 5;
  const int lane   = threadIdx.x & 31;
  const int tilesN = Nn >> 4;
  const int T      = (M >> 4) * tilesN;
  const int t      = blockIdx.x * 4 + wave;
  if (t >= T) return;
  const int mt = t / tilesN;
  const int nt = t - mt * tilesN;
  const int m0 = mt << 4, n0 = nt << 4;
  const int z  = blockIdx.z;
  const float* Ab = A + (long)z * sA;
  const float* Bb = B + (long)z * sB;
  const int am = m0 + (lane & 15);
  const int bn = n0 + (lane & 15);
  const int kh = (lane >> 4) << 1;

  v8f c = {0.f,0.f,0.f,0.f,0.f,0.f,0.f,0.f};
  const float* arow = Ab + (long)am * K + kh;
  const float* bcol = Bb + (long)kh * Nn + bn;
  for (int k0 = 0; k0 < K; k0 += 4) {
    v2f a, b;
    a.x = arow[0]; a.y = arow[1];
    b.x = bcol[0]; b.y = bcol[Nn];
    arow += 4;
    bcol += (long)4 * Nn;
    c = __builtin_amdgcn_wmma_f32_16x16x4_f32(false, a, false, b, (short)0, c, false, false);
  }

  const float bv = bias ? bias[(long)z * sBias + bn] : 0.f;
  float* Db = D + (long)z * sD;
  const int rbase = m0 + ((lane >> 4) << 3);
#pragma unroll
  for (int v = 0; v < 8; ++v) {
    float x = c[v] + bv;
    if (ACT == 1)      x = fmaxf(x, 0.f);
    else if (ACT == 2) x = (x > 0.f) ? x : 0.01f * x;
    else if (ACT == 3) x = tanhf(x);
    Db[(long)(rbase + v) * Nn + bn] = x;
  }
}

// ---------------------------------------------------------------------------
__global__ void fill_zero_f32(float* __restrict__ p, long n) {
  long i = (long)blockIdx.x * blockDim.x + threadIdx.x;
  long stride = (long)gridDim.x * blockDim.x;
  for (; i < n; i += stride) p[i] = 0.f;
}

__global__ void count_deg(const int* __restrict__ edges, float* __restrict__ deg) {
  long i = (long)blockIdx.x * blockDim.x + threadIdx.x;
  if (i >= (long)R_REL * E_EDGES) return;
  int r   = (int)(i / E_EDGES);
  int dst = edges[i * 2 + 1];
  atomicAdd(&deg[(long)r * N_NODES + dst], 1.0f);
}

__global__ void inv_deg(float* __restrict__ deg) {
  long i = (long)blockIdx.x * blockDim.x + threadIdx.x;
  if (i >= (long)R_REL * N_NODES) return;
  deg[i] = 1.0f / fmaxf(deg[i], 1.0f);
}

__global__ void compute_wr(const float* __restrict__ C, const float* __restrict__ V,
                           float* __restrict__ Wr) {
  int i = blockIdx.x * blockDim.x + threadIdx.x;
  if (i >= R_REL * HD * HD) return;
  int r  = i >> 12;
  int io = i & (HD * HD - 1);
  Wr[i] = C[r * 2] * V[io] + C[r * 2 + 1] * V[HD * HD + io];
}

__global__ __launch_bounds__(256) void scatter_edges(
    const float* __restrict__ m, const int* __restrict__ edges,
    const float* __restrict__ degInv, float* __restrict__ out)
{
  const int lane = threadIdx.x & 31;
  const int e    = blockIdx.x * 8 + (threadIdx.x >> 5);
  const int r    = blockIdx.y;
  if (e >= E_EDGES) return;
  const int2  ed = ((const int2*)edges)[(long)r * E_EDGES + e];
  const float w  = degInv[(long)r * N_NODES + ed.y];
  const float2 mv = ((const float2*)(m + ((long)r * N_NODES + ed.x) * HD))[lane];
  float* o = out + (long)ed.y * HD + lane * 2;
  atomicAdd(o,     mv.x * w);
  atomicAdd(o + 1, mv.y * w);
}

__global__ void relu_inplace(float* __restrict__ a, long n) {
  long i = (long)blockIdx.x * blockDim.x + threadIdx.x;
  long stride = (long)gridDim.x * blockDim.x;
  for (; i < n; i += stride) a[i] = fmaxf(a[i], 0.f);
}

__global__ void add_inplace(float* __restrict__ a, const float* __restrict__ b, long n) {
  long i = (long)blockIdx.x * blockDim.x + threadIdx.x;
  long stride = (long)gridDim.x * blockDim.x;
  for (; i < n; i += stride) a[i] += b[i];
}

__global__ void gather_rows(const float* __restrict__ h, const int* __restrict__ seeds,
                            float* __restrict__ hs) {
  int i = blockIdx.x * blockDim.x + threadIdx.x;
  if (i >= S_SEEDS * HD) return;
  int s = i >> 6, c = i & 63;
  hs[i] = h[(long)seeds[s] * HD + c];
}

__global__ void head_d3(const float* __restrict__ d2, const float* __restrict__ Wd3,
                        const float* __restrict__ bd3, float* __restrict__ out) {
  int i = blockIdx.x * blockDim.x + threadIdx.x;
  if (i >= P_HEADS * S_SEEDS) return;
  int p = i >> 12;
  const float* v = d2 + (long)i * 32;
  const float* w = Wd3 + p * 32;
  float acc = bd3[p];
#pragma unroll
  for (int k = 0; k < 32; ++k) acc += v[k] * w[k];
  out[i] = fmaxf(acc, 0.f);
}

// ---------------------------------------------------------------------------
extern "C" void kernel_launch(void* const* d_in, const int* in_sizes, int n_in,
                              void* d_out, int out_size, void* d_ws, size_t ws_size,
                              hipStream_t stream)
{
  const float* x     = (const float*)d_in[0];
  const int*   edges = (const int*)  d_in[1];
  const int*   seeds = (const int*)  d_in[2];
  const float* noise = (const float*)d_in[3];
  const float* V1    = (const float*)d_in[4];
  const float* C1    = (const float*)d_in[5];
  const float* V2    = (const float*)d_in[6];
  const float* C2    = (const float*)d_in[7];
  const float* W_lin = (const float*)d_in[8];
  const float* b_lin = (const float*)d_in[9];
  const float* Wd1   = (const float*)d_in[10];
  const float* bd1   = (const float*)d_in[11];
  const float* Wd2   = (const float*)d_in[12];
  const float* bd2   = (const float*)d_in[13];
  const float* Wd3   = (const float*)d_in[14];
  const float* bd3   = (const float*)d_in[15];
  const float* Wf1   = (const float*)d_in[16];
  const float* bf1   = (const float*)d_in[17];
  const float* Wf2   = (const float*)d_in[18];
  const float* bf2   = (const float*)d_in[19];
  const float* Wf3   = (const float*)d_in[20];
  const float* bf3   = (const float*)d_in[21];
  float* out = (float*)d_out;

  // ---- workspace layout (floats) ----
  float* ws     = (float*)d_ws;
  float* degInv = ws;                                   // R*N
  float* wr     = degInv + (long)R_REL * N_NODES;       // R*64*64
  float* mbuf   = wr + R_REL * HD * HD;                 // R*N*64
  float* h1     = mbuf + (long)R_REL * N_NODES * HD;    // N*64
  float* h2     = h1 + (long)N_NODES * HD;              // N*64
  float* hs     = h2 + (long)N_NODES * HD;              // S*64
  float* d1 = mbuf;                                     // alias m region
  float* d2 = d1 + (long)P_HEADS * S_SEEDS * 256;
  float* f1 = d2 + (long)P_HEADS * S_SEEDS * 32;
  float* f2 = f1 + (long)P_HEADS * S_SEEDS * 256;
  float* h3 = h1;

  const int TB = 256;
  const long NH = (long)N_NODES * HD;
  const int MtilesN = N_NODES / 16;                     // 6250
  const int gxN = (MtilesN + 3) / 4;                    // 1563

  // ---- inverse degrees ----
  fill_zero_f32<<<1024, TB, 0, stream>>>(degInv, (long)R_REL * N_NODES);
  count_deg<<<(R_REL * E_EDGES + TB - 1) / TB, TB, 0, stream>>>(edges, degInv);
  inv_deg<<<(R_REL * N_NODES + TB - 1) / TB, TB, 0, stream>>>(degInv);

  // ---- conv layer 1 ----
  compute_wr<<<(R_REL * HD * HD) / TB, TB, 0, stream>>>(C1, V1, wr);
  gemm_wmma_tiled<0><<<dim3(gxN, 1, R_REL), 128, 0, stream>>>(
      x, wr, nullptr, mbuf, N_NODES, HD, FEAT, 0L, (long)HD * HD, 0L, NH);
  fill_zero_f32<<<1024, TB, 0, stream>>>(h1, NH);
  scatter_edges<<<dim3((E_EDGES + 7) / 8, R_REL), 256, 0, stream>>>(mbuf, edges, degInv, h1);
  relu_inplace<<<1024, TB, 0, stream>>>(h1, NH);

  // ---- conv layer 2 ----
  compute_wr<<<(R_REL * HD * HD) / TB, TB, 0, stream>>>(C2, V2, wr);
  gemm_wmma_tiled<0><<<dim3(gxN, 1, R_REL), 128, 0, stream>>>(
      h1, wr, nullptr, mbuf, N_NODES, HD, HD, 0L, (long)HD * HD, 0L, NH);
  fill_zero_f32<<<1024, TB, 0, stream>>>(h2, NH);
  scatter_edges<<<dim3((E_EDGES + 7) / 8, R_REL), 256, 0, stream>>>(mbuf, edges, degInv, h2);
  relu_inplace<<<1024, TB, 0, stream>>>(h2, NH);

  // ---- h3 = leaky_relu(h2 @ W_lin + b_lin) + noise ----
  gemm_wmma_tiled<2><<<dim3(gxN, 1, 1), 128, 0, stream>>>(
      h2, W_lin, b_lin, h3, N_NODES, HD, HD, 0L, 0L, 0L, 0L);
  add_inplace<<<1024, TB, 0, stream>>>(h3, noise, NH);
  gather_rows<<<(S_SEEDS * HD) / TB, TB, 0, stream>>>(h3, seeds, hs);

  // ---- missing-node head: 64 -> 256 -> 32 -> 1 ----
  gemm_wmma_tiled<2><<<dim3(S_SEEDS / 64, 256 / 64, P_HEADS), 128, 0, stream>>>(
      hs, Wd1, bd1, d1, S_SEEDS, 256, HD,
      0L, (long)HD * 256, 256L, (long)S_SEEDS * 256);
  {
    const int T = (S_SEEDS / 16) * (32 / 16);
    gemm_wmma_small<2><<<dim3((T + 3) / 4, 1, P_HEADS), 128, 0, stream>>>(
        d1, Wd2, bd2, d2, S_SEEDS, 32, 256,
        (long)S_SEEDS * 256, 256L * 32, 32L, (long)S_SEEDS * 32);
  }
  head_d3<<<(P_HEADS * S_SEEDS) / TB, TB, 0, stream>>>(d2, Wd3, bd3, out);

  // ---- feature head: 64 -> 256 -> 2048 -> 320(tanh) ----
  gemm_wmma_tiled<1><<<dim3(S_SEEDS / 64, 256 / 64, P_HEADS), 128, 0, stream>>>(
      hs, Wf1, bf1, f1, S_SEEDS, 256, HD,
      0L, (long)HD * 256, 256L, (long)S_SEEDS * 256);
  gemm_wmma_tiled<1><<<dim3(S_SEEDS / 64, 2048 / 64, P_HEADS), 128, 0, stream>>>(
      f1, Wf2, bf2, f2, S_SEEDS, 2048, 256,
      (long)S_SEEDS * 256, 256L * 2048, 2048L, (long)S_SEEDS * 2048);
  gemm_wmma_tiled<3><<<dim3(S_SEEDS / 64, 320 / 64, P_HEADS), 128, 0, stream>>>(
      f2, Wf3, bf3, out + (long)P_HEADS * S_SEEDS, S_SEEDS, 320, 2048,
      (long)S_SEEDS * 2048, 2048L * 320, 320L, (long)S_SEEDS * 320);
}